// FilterTransform_4415226380424
// MI455X (gfx1250) — compile-verified
//
#include <hip/hip_runtime.h>

typedef __attribute__((ext_vector_type(2))) float v2f;
typedef __attribute__((ext_vector_type(4))) float v4f;
typedef __attribute__((ext_vector_type(8))) float v8f;

#define T_LEN 16000
#define B_ROWS 4096
#define BLK 16
#define NBLK (T_LEN / BLK)

// ---------------------------------------------------------------------------
// Prep: compose HP biquad x second-stage biquad into one 4th-order IIR, then
// build block matrices for the direct-form-I block recurrence:
//   y[m] = Sum_k g4[m-k] x[k] + Sum_i cx_i[m] x_{-i} + Sum_i cy_i[m] y_{-i}
// g4  = full impulse response of the cascade (first 16 taps)
// gA4 = impulse response of 1/A'(z)
// cx_i[m] = Sum_{k=0}^{4-i} B'_{k+i} gA4[m-k]
// cy_i[m] = -Sum_{j=i}^{4}  A'_j    gA4[m+i-j]
// ws layout per filter f (0 = x-signal, 1 = n-signal), 384 floats each:
//   Amain[16][16] (Toeplitz of g4), Ahx[16][4] (cols x_{-4}..x_{-1}),
//   Ahy[16][4] (cols y_{-4}..y_{-1}).
// ---------------------------------------------------------------------------
__global__ void prep_matrices(const float* __restrict__ a_x, const float* __restrict__ b_x,
                              const float* __restrict__ a_n, const float* __restrict__ b_n,
                              float* __restrict__ ws) {
    int f = threadIdx.x;
    if (f >= 2) return;
    // stage 1: fixed high-pass
    const float b0 = -2.0f, b1 = 1.0f, a0 = -1.99599f, a1 = 0.996f;
    // stage 2: runtime coefficients (first two entries of the batch vectors)
    const float* bb = f ? b_n : b_x;
    const float* aa = f ? a_n : a_x;
    const float c0 = bb[0], c1 = bb[1], d0 = aa[0], d1 = aa[1];

    // composed 4th-order polynomials (index 0..4)
    float Bc[5] = {1.0f, b0 + c0, b1 + b0 * c0 + c1, b0 * c1 + b1 * c0, b1 * c1};
    float Ac[5] = {1.0f, a0 + d0, a1 + a0 * d0 + d1, a0 * d1 + a1 * d0, a1 * d1};

    float gA[16];
    for (int m = 0; m < 16; ++m) {
        float s = (m == 0) ? 1.0f : 0.0f;
        for (int j = 1; j <= 4; ++j)
            if (m - j >= 0) s -= Ac[j] * gA[m - j];
        gA[m] = s;
    }
    float g4[16];
    for (int m = 0; m < 16; ++m) {
        float s = 0.0f;
        for (int i = 0; i <= 4; ++i)
            if (m - i >= 0) s += Bc[i] * gA[m - i];
        g4[m] = s;
    }

    float* Am  = ws + f * 384;
    float* Ahx = Am + 256;
    float* Ahy = Ahx + 64;
    for (int m = 0; m < 16; ++m) {
        for (int k = 0; k < 16; ++k) Am[m * 16 + k] = (k <= m) ? g4[m - k] : 0.0f;
        for (int j = 0; j < 4; ++j) {        // column j corresponds to lag i = 4-j
            int i = 4 - j;
            float sx = 0.0f;
            for (int k = 0; k + i <= 4; ++k)
                if (m - k >= 0) sx += Bc[k + i] * gA[m - k];
            Ahx[m * 4 + j] = sx;
            float sy = 0.0f;
            for (int jj = i; jj <= 4; ++jj)
                if (m + i - jj >= 0) sy -= Ac[jj] * gA[m + i - jj];
            Ahy[m * 4 + j] = sy;
        }
    }
}

// ---------------------------------------------------------------------------
// Main: one wave per (signal, 16-row batch tile); 1000 blocks of 16 steps.
// Per block: 6x V_WMMA_F32_16X16X4_F32. The x-history operand is literally
// the previous block's Bx[3] register (lane layout matches); the y-history
// operand needs only two cross-half shuffles.
// A operand (16x4): lane m holds A[m][2*half+j]. B operand (4x16): lane n
// holds B[2*half+j][n]. D (16x16): VGPR r = row r+8*half, col = lane&15.
// ---------------------------------------------------------------------------
__global__ __launch_bounds__(256) void iir_wmma_kernel(const float* __restrict__ x,
                                                       const float* __restrict__ nn,
                                                       const float* __restrict__ ws,
                                                       float* __restrict__ out) {
    const int lane = threadIdx.x & 31;
    const int wave = threadIdx.x >> 5;
    const int gw   = blockIdx.x * 8 + wave;   // 0..511
    const int sig  = gw >> 8;                 // 0 = x, 1 = n
    const int tile = gw & 255;
    const int row0 = tile * 16;
    const int col  = lane & 15;
    const int hf   = lane >> 4;

    const float* in  = sig ? nn : x;
    float*       o   = out + (size_t)sig * (size_t)B_ROWS * (size_t)T_LEN;
    const float* src = in + (size_t)(row0 + col) * (size_t)T_LEN;
    float*       dst = o  + (size_t)(row0 + col) * (size_t)T_LEN;

    const float* Am  = ws + (size_t)sig * 384;
    const float* Ahx = Am + 256;
    const float* Ahy = Ahx + 64;

    v2f aU[4];
#pragma unroll
    for (int kc = 0; kc < 4; ++kc)
        aU[kc] = *(const v2f*)(Am + col * 16 + 4 * kc + 2 * hf);
    v2f aHx = *(const v2f*)(Ahx + col * 4 + 2 * hf);
    v2f aHy = *(const v2f*)(Ahy + col * 4 + 2 * hf);

    // carried history (fresh zero state per reference)
    v2f   pB3 = {0.0f, 0.0f};                 // prev x[12..15] (B layout)
    float py12 = 0.0f, py13 = 0.0f;           // prev y[12],y[13] (lanes 16-31)
    float py14 = 0.0f, py15 = 0.0f;           // prev y[14],y[15] (lanes 16-31)

    for (int blk = 0; blk < NBLK; ++blk) {
        const int t0 = blk * BLK;

        v2f Bx[4];
#pragma unroll
        for (int kc = 0; kc < 4; ++kc)
            Bx[kc] = *(const v2f*)(src + t0 + 4 * kc + 2 * hf);

        // branchless clamped prefetch ~16 blocks ahead
        int pf = t0 + 256;
        pf = (pf > T_LEN - BLK) ? (T_LEN - BLK) : pf;
        __builtin_prefetch(src + pf, 0, 3);

        // main Toeplitz: y = U @ x_block
        v8f y = {};
#pragma unroll
        for (int kc = 0; kc < 4; ++kc)
            y = __builtin_amdgcn_wmma_f32_16x16x4_f32(false, aU[kc], false, Bx[kc],
                                                      (short)0, y, false, false);

        // x-history: previous Bx[3] already has rows [x_{-4}..x_{-1}] in B layout
        y = __builtin_amdgcn_wmma_f32_16x16x4_f32(false, aHx, false, pB3,
                                                  (short)0, y, false, false);

        // y-history: rows [y_{-4}..y_{-1}] = prev y[12..15]
        float s12 = __shfl_xor(py12, 16);     // y[12] -> lanes 0-15
        float s13 = __shfl_xor(py13, 16);     // y[13] -> lanes 0-15
        v2f hy;
        hy.x = hf ? py14 : s12;               // K0: y_{-4} | K2: y_{-2}
        hy.y = hf ? py15 : s13;               // K1: y_{-3} | K3: y_{-1}
        y = __builtin_amdgcn_wmma_f32_16x16x4_f32(false, aHy, false, hy,
                                                  (short)0, y, false, false);

        // carry history
        pB3  = Bx[3];
        py12 = y[4]; py13 = y[5]; py14 = y[6]; py15 = y[7];

        // store: lane holds times t0+8*hf+0..7 of its row, two b128 stores
        v4f lo = {y[0], y[1], y[2], y[3]};
        v4f hi = {y[4], y[5], y[6], y[7]};
        *(v4f*)(dst + t0 + 8 * hf)     = lo;
        *(v4f*)(dst + t0 + 8 * hf + 4) = hi;
    }
}

extern "C" void kernel_launch(void* const* d_in, const int* in_sizes, int n_in,
                              void* d_out, int out_size, void* d_ws, size_t ws_size,
                              hipStream_t stream) {
    const float* x   = (const float*)d_in[0];
    const float* n   = (const float*)d_in[1];
    const float* a_x = (const float*)d_in[2];
    const float* b_x = (const float*)d_in[3];
    const float* a_n = (const float*)d_in[4];
    const float* b_n = (const float*)d_in[5];
    float* ws  = (float*)d_ws;
    float* out = (float*)d_out;

    prep_matrices<<<1, 32, 0, stream>>>(a_x, b_x, a_n, b_n, ws);
    // 512 waves = 2 signals x 256 batch tiles; 8 waves per block
    iir_wmma_kernel<<<64, 256, 0, stream>>>(x, n, ws, out);
}